// BAMPLayer_55783035240673
// MI455X (gfx1250) — compile-verified
//
#include <hip/hip_runtime.h>
#include <hip/hip_bf16.h>
#include <math.h>

// Problem constants
#define B_    8
#define NR    4096
#define LM_   8192
#define LSEG  128
#define MSEG  64
#define KSYM  4
#define SIGMA2 0.01f

typedef float v2f __attribute__((ext_vector_type(2)));
typedef float v4f __attribute__((ext_vector_type(4)));
typedef float v8f __attribute__((ext_vector_type(8)));

static __device__ __forceinline__ v8f wmma4(v2f a, v2f b, v8f c) {
  // D = A(16x4 f32) * B(4x16 f32) + C(16x16 f32)
  return __builtin_amdgcn_wmma_f32_16x16x4_f32(
      /*neg_a=*/false, a, /*neg_b=*/false, b,
      /*c_mod=*/(short)0, c, /*reuse_a=*/false, /*reuse_b=*/false);
}

// ---------------- workspace layout (floats) ----------------
// P1[col][pair] = float4{ Bx(2p,col), Bx(2p+1,col), Bv(2p,col), Bv(2p+1,col) }
//   col 0..7  -> Bx = x_re[b=col], Bv = var[b=col]
//   col 8..15 -> Bx = x_im[b=col-8], Bv = 0
#define WS_P1   0                          // 16 * (LM/2) * 4 = LM*32
// P2[col][pair] = float4{ Br(2p,col), Br(2p+1,col), Bu(2p,col), Bu(2p+1,col) }
#define WS_P2   (WS_P1 + LM_*32)           // 16 * (NR/2) * 4 = NR*32
#define WS_ZR   (WS_P2 + NR*32)            // B*NR   z_new re
#define WS_ZI   (WS_ZR + B_*NR)            // B*NR   z_new im
#define WS_UN   (WS_ZI + B_*NR)            // B*NR   u_new
#define WS_XR   (WS_UN + B_*NR)            // B*LM   xmap re
#define WS_XI   (WS_XR + B_*LM_)           // B*LM   xmap im
#define WS_CV   (WS_XI + B_*LM_)           // B*LM   cov
#define WS_X    (WS_CV + B_*LM_)           // B*LM*4 denoiser x values
#define WS_GM   (WS_X  + B_*LM_*KSYM)      // 1      global max (uint bits)

// ---------------- output layout (floats) ----------------
#define OUT_XM  0                          // B*LM complex interleaved
#define OUT_VR  (2*B_*LM_)                 // B*LM
#define OUT_ZN  (OUT_VR + B_*LM_)          // B*NR complex interleaved
#define OUT_UN  (OUT_ZN + 2*B_*NR)         // B*NR

// Pack B-operands for pass 1 (one float4 per (col, k-pair))
__global__ void prep1_kernel(const float* __restrict__ xre,
                             const float* __restrict__ xim,
                             const float* __restrict__ var,
                             float* __restrict__ ws) {
  int tid = blockIdx.x * blockDim.x + threadIdx.x;   // 16 * LM/2 threads
  if (tid == 0) ((unsigned*)(ws + WS_GM))[0] = 0u;
  if (tid >= 16 * (LM_ / 2)) return;
  int c = tid >> 12;              // column 0..15
  int p = tid & (LM_ / 2 - 1);    // pair 0..4095
  int m = 2 * p;
  v4f q;
  if (c < 8) {
    q.x = xre[c * LM_ + m];
    q.y = xre[c * LM_ + m + 1];
    q.z = var[c * LM_ + m];
    q.w = var[c * LM_ + m + 1];
  } else {
    q.x = xim[(c - 8) * LM_ + m];
    q.y = xim[(c - 8) * LM_ + m + 1];
    q.z = 0.0f;
    q.w = 0.0f;
  }
  ((v4f*)(ws + WS_P1))[tid] = q;
}

// Pass 1: rows of H. v = |H|^2 var ; Hx = H x ; z_new, u_new
__global__ void __launch_bounds__(256)
pass1_kernel(const float* __restrict__ Hre, const float* __restrict__ Him,
             const float* __restrict__ yre, const float* __restrict__ yim,
             const float* __restrict__ zre, const float* __restrict__ zim,
             const float* __restrict__ uin,
             float* __restrict__ ws, float* __restrict__ out) {
  const int lane = threadIdx.x & 31;
  const int wave = threadIdx.x >> 5;
  const int n0   = blockIdx.x * 16;
  const int half = lane >> 4;          // 0: K=0,1 / M-lo ; 1: K=2,3 / M-hi
  const int lo   = lane & 15;
  const int cb   = half * 2;

  const float* pAre = Hre + (size_t)(n0 + lo) * LM_ + cb;
  const float* pAim = Him + (size_t)(n0 + lo) * LM_ + cb;
  const v4f*   pP   = (const v4f*)(ws + WS_P1) + (size_t)lo * (LM_ / 2) + half;

  v8f acc1 = {0}, acc2 = {0}, acc3 = {0};
  const int k0 = wave * (LM_ / 8);
#pragma unroll 4
  for (int k = k0; k < k0 + (LM_ / 8); k += 4) {
    v2f are = *(const v2f*)(pAre + k);
    v2f aim = *(const v2f*)(pAim + k);
    v2f aab = are * are + aim * aim;
    v4f q   = pP[k >> 1];                 // {bx0,bx1,bv0,bv1}
    v2f bx; bx.x = q.x; bx.y = q.y;
    v2f bv; bv.x = q.z; bv.y = q.w;
    acc1 = wmma4(are, bx, acc1);   // H_re * [x_re | x_im]
    acc2 = wmma4(aim, bx, acc2);   // H_im * [x_re | x_im]
    acc3 = wmma4(aab, bv, acc3);   // |H|^2 * [var | 0]
  }

  // deterministic cross-wave reduction (fixed order)
  __shared__ float red[8][32][24];
#pragma unroll
  for (int i = 0; i < 8; ++i) {
    red[wave][lane][i]      = acc1[i];
    red[wave][lane][8 + i]  = acc2[i];
    red[wave][lane][16 + i] = acc3[i];
  }
  __syncthreads();
  if (wave == 0) {
    for (int w = 1; w < 8; ++w)
#pragma unroll
      for (int i = 0; i < 8; ++i) {
        acc1[i] += red[w][lane][i];
        acc2[i] += red[w][lane][8 + i];
        acc3[i] += red[w][lane][16 + i];
      }
    float hxre[8], hxim[8], vv[8];
#pragma unroll
    for (int j = 0; j < 8; ++j) {
      float a1h = __shfl_down(acc1[j], 8, 32);  // column b+8 (H_re*x_im)
      float a2h = __shfl_down(acc2[j], 8, 32);  // column b+8 (H_im*x_im)
      hxre[j] = acc1[j] - a2h;
      hxim[j] = a1h + acc2[j];
      vv[j]   = acc3[j];
    }
    if (lo < 8) {
      const int b = lo;
#pragma unroll
      for (int j = 0; j < 8; ++j) {
        int n = n0 + j + half * 8;
        size_t idx = (size_t)b * NR + n;
        float un  = vv[j] + SIGMA2;
        float s   = vv[j] / uin[idx];
        float znr = hxre[j] - s * (yre[idx] - zre[idx]);
        float zni = hxim[j] - s * (yim[idx] - zim[idx]);
        out[OUT_ZN + 2 * idx]     = znr;
        out[OUT_ZN + 2 * idx + 1] = zni;
        out[OUT_UN + idx]         = un;
        ws[WS_ZR + idx] = znr;
        ws[WS_ZI + idx] = zni;
        ws[WS_UN + idx] = un;
      }
    }
  }
}

// Pack B-operands for pass 2 (one float4 per (col, n-pair))
__global__ void prep2_kernel(const float* __restrict__ yre,
                             const float* __restrict__ yim,
                             float* __restrict__ ws) {
  int tid = blockIdx.x * blockDim.x + threadIdx.x;   // 16 * NR/2 threads
  if (tid >= 16 * (NR / 2)) return;
  int c = tid >> 11;              // column 0..15
  int p = tid & (NR / 2 - 1);     // pair 0..2047
  int n = 2 * p;
  int b = (c < 8) ? c : c - 8;
  size_t i0 = (size_t)b * NR + n;
  size_t i1 = i0 + 1;
  float inv0 = 1.0f / ws[WS_UN + i0];
  float inv1 = 1.0f / ws[WS_UN + i1];
  v4f q;
  if (c < 8) {
    q.x = (yre[i0] - ws[WS_ZR + i0]) * inv0;
    q.y = (yre[i1] - ws[WS_ZR + i1]) * inv1;
    q.z = inv0;
    q.w = inv1;
  } else {
    q.x = (yim[i0] - ws[WS_ZI + i0]) * inv0;
    q.y = (yim[i1] - ws[WS_ZI + i1]) * inv1;
    q.z = 0.0f;
    q.w = 0.0f;
  }
  ((v4f*)(ws + WS_P2))[tid] = q;
}

// Pass 2: columns of H. covsum = |H|^2^T (1/u) ; H^H r ; cov, xmap
__global__ void __launch_bounds__(256)
pass2_kernel(const float* __restrict__ Hre, const float* __restrict__ Him,
             const float* __restrict__ xre, const float* __restrict__ xim,
             float* __restrict__ ws) {
  const int lane = threadIdx.x & 31;
  const int wave = threadIdx.x >> 5;
  const int m0   = blockIdx.x * 16;
  const int half = lane >> 4;
  const int lo   = lane & 15;
  const int rb   = half * 2;

  const float* pAre = Hre + (m0 + lo);
  const float* pAim = Him + (m0 + lo);
  const v4f*   pQ   = (const v4f*)(ws + WS_P2) + (size_t)lo * (NR / 2) + half;

  v8f acc1 = {0}, acc2 = {0}, acc3 = {0};
  const int k0 = wave * (NR / 8);
#pragma unroll 4
  for (int k = k0; k < k0 + (NR / 8); k += 4) {
    size_t r0 = (size_t)(k + rb) * LM_;
    v2f are; are.x = pAre[r0]; are.y = pAre[r0 + LM_];
    v2f aim; aim.x = pAim[r0]; aim.y = pAim[r0 + LM_];
    v2f aab = are * are + aim * aim;
    v4f q   = pQ[k >> 1];                 // {br0,br1,bu0,bu1}
    v2f br; br.x = q.x; br.y = q.y;
    v2f bu; bu.x = q.z; bu.y = q.w;
    acc1 = wmma4(are, br, acc1);   // H_re^T * [r_re | r_im]
    acc2 = wmma4(aim, br, acc2);   // H_im^T * [r_re | r_im]
    acc3 = wmma4(aab, bu, acc3);   // |H|^2^T * [1/u | 0]
  }

  __shared__ float red[8][32][24];
#pragma unroll
  for (int i = 0; i < 8; ++i) {
    red[wave][lane][i]      = acc1[i];
    red[wave][lane][8 + i]  = acc2[i];
    red[wave][lane][16 + i] = acc3[i];
  }
  __syncthreads();
  if (wave == 0) {
    for (int w = 1; w < 8; ++w)
#pragma unroll
      for (int i = 0; i < 8; ++i) {
        acc1[i] += red[w][lane][i];
        acc2[i] += red[w][lane][8 + i];
        acc3[i] += red[w][lane][16 + i];
      }
    float hrre[8], hrim[8], cs[8];
#pragma unroll
    for (int j = 0; j < 8; ++j) {
      float a1h = __shfl_down(acc1[j], 8, 32);  // H_re^T r_im
      float a2h = __shfl_down(acc2[j], 8, 32);  // H_im^T r_im
      hrre[j] = acc1[j] + a2h;   // Re(H^H r) = Hre*rre + Him*rim
      hrim[j] = a1h - acc2[j];   // Im(H^H r) = Hre*rim - Him*rre
      cs[j]   = acc3[j];
    }
    if (lo < 8) {
      const int b = lo;
#pragma unroll
      for (int j = 0; j < 8; ++j) {
        int m = m0 + j + half * 8;
        size_t idx = (size_t)b * LM_ + m;
        float cov = 1.0f / cs[j];
        ws[WS_CV + idx] = cov;
        ws[WS_XR + idx] = xre[idx] + cov * hrre[j];
        ws[WS_XI + idx] = xim[idx] + cov * hrim[j];
      }
    }
  }
}

// Denoiser stage A: x values + global max|x| (order-independent -> deterministic)
__global__ void denoiseA_kernel(const float* __restrict__ symr,
                                const float* __restrict__ symi,
                                float* __restrict__ ws) {
  int idx = blockIdx.x * blockDim.x + threadIdx.x;   // b*LM + m
  if (idx >= B_ * LM_) return;
  float sr = ws[WS_XR + idx], si = ws[WS_XI + idx];
  float inv_t = 1.0f / (ws[WS_CV + idx] * 0.5f);
  float amax = 0.0f;
#pragma unroll
  for (int k = 0; k < KSYM; ++k) {
    float x = (sr * symr[k] + si * symi[k]) * inv_t;  // Re((s/tau)*conj(sym))
    ws[WS_X + (size_t)idx * KSYM + k] = x;
    amax = fmaxf(amax, fabsf(x));
  }
#pragma unroll
  for (int o = 16; o > 0; o >>= 1) amax = fmaxf(amax, __shfl_xor(amax, o, 32));
  if ((threadIdx.x & 31) == 0)
    atomicMax((unsigned*)(ws + WS_GM), __float_as_uint(amax));
}

// Denoiser stage B: segmented softmax over (M=64, K=4) per (b,l)
__global__ void __launch_bounds__(64)
denoiseB_kernel(const float* __restrict__ symr, const float* __restrict__ symi,
                const float* __restrict__ ws, float* __restrict__ out) {
  const int b = blockIdx.x >> 7;        // /LSEG
  const int l = blockIdx.x & (LSEG - 1);
  const int m = threadIdx.x;            // 0..63
  const size_t idx = (size_t)b * LM_ + l * MSEG + m;
  const float gmax = __uint_as_float(((const unsigned*)(ws + WS_GM))[0]);

  float eta[KSYM], sm = 0.0f;
#pragma unroll
  for (int k = 0; k < KSYM; ++k) {
    eta[k] = expf(ws[WS_X + idx * KSYM + k] - gmax);
    sm += eta[k];
  }
  __shared__ float sden[MSEG];
  sden[m] = sm;
  __syncthreads();
  for (int s = MSEG / 2; s > 0; s >>= 1) {
    if (m < s) sden[m] += sden[m + s];
    __syncthreads();
  }
  float inv_den = 1.0f / sden[0];
  float numr = 0.0f, numi = 0.0f, pwr = 0.0f;
#pragma unroll
  for (int k = 0; k < KSYM; ++k) {
    numr += symr[k] * eta[k];
    numi += symi[k] * eta[k];
    pwr  += (symr[k] * symr[k] + symi[k] * symi[k]) * eta[k];
  }
  float xmr = numr * inv_den, xmi = numi * inv_den;
  out[OUT_XM + 2 * idx]     = xmr;
  out[OUT_XM + 2 * idx + 1] = xmi;
  out[OUT_VR + idx]         = pwr * inv_den - (xmr * xmr + xmi * xmi);
}

extern "C" void kernel_launch(void* const* d_in, const int* in_sizes, int n_in,
                              void* d_out, int out_size, void* d_ws, size_t ws_size,
                              hipStream_t stream) {
  const float* Hre  = (const float*)d_in[0];
  const float* Him  = (const float*)d_in[1];
  const float* yre  = (const float*)d_in[2];
  const float* yim  = (const float*)d_in[3];
  const float* xre  = (const float*)d_in[4];
  const float* xim  = (const float*)d_in[5];
  const float* var  = (const float*)d_in[6];
  const float* zre  = (const float*)d_in[7];
  const float* zim  = (const float*)d_in[8];
  const float* uin  = (const float*)d_in[9];
  const float* symr = (const float*)d_in[10];
  const float* symi = (const float*)d_in[11];
  float* out = (float*)d_out;
  float* ws  = (float*)d_ws;

  prep1_kernel<<<(16 * (LM_ / 2)) / 256, 256, 0, stream>>>(xre, xim, var, ws);
  pass1_kernel<<<NR / 16, 256, 0, stream>>>(Hre, Him, yre, yim, zre, zim, uin, ws, out);
  prep2_kernel<<<(16 * (NR / 2)) / 256, 256, 0, stream>>>(yre, yim, ws);
  pass2_kernel<<<LM_ / 16, 256, 0, stream>>>(Hre, Him, xre, xim, ws);
  denoiseA_kernel<<<(B_ * LM_) / 256, 256, 0, stream>>>(symr, symi, ws);
  denoiseB_kernel<<<B_ * LSEG, MSEG, 0, stream>>>(symr, symi, ws, out);
}